// EctLayer_1803886264527
// MI455X (gfx1250) — compile-verified
//
#include <hip/hip_runtime.h>

// Problem constants (from the reference): N=20000, E=40000, B=32, S=32, T=64, F=3
#define T_THETA 64
#define S_STEPS 32
#define B_GRAPH 32
#define N_CHUNK 8

typedef __attribute__((ext_vector_type(2))) float v2f;
typedef __attribute__((ext_vector_type(8))) float v8f;

// ---------------------------------------------------------------------------
// Kernel 1: nh[N,64] = x[N,3] @ v[3,64] via V_WMMA_F32_16X16X4_F32 (K padded 3->4).
// One wave computes a 16(M) x 64(T) tile with 4 WMMA ops.
// A (16x4 f32) layout, ISA 7.12.2: lanes 0-15: VGPR0=K0, VGPR1=K1;
//                                  lanes 16-31: VGPR0=K2, VGPR1=K3.
// B (4x16 f32) mirrors A; D = standard 8-VGPR 16x16 f32 C/D layout.
// Loads use clamped addresses + selects (no divergent branches), stores use a
// wave-uniform full-tile fast path so the hot path is 32 plain b32 stores.
// ---------------------------------------------------------------------------
__global__ void nh_gemm_wmma(const float* __restrict__ x,
                             const float* __restrict__ v,
                             float* __restrict__ nh, int n) {
  const int gwave   = (blockIdx.x * blockDim.x + threadIdx.x) >> 5;
  const int lane    = threadIdx.x & 31;
  const int rowBase = gwave * 16;
  if (rowBase >= n) return;          // wave-uniform: whole wave exits or none

  const int m  = lane & 15;
  const int kh = lane >> 4;          // 0 -> K={0,1}, 1 -> K={2,3}
  const int rm = rowBase + m;
  const int rc = (rm < n) ? rm : (n - 1);   // clamped row for loads

  // Unconditional (clamped) loads + selects: keeps EXEC all-1s into the WMMA.
  const float xa = x[rc * 3 + (kh << 1)];   // K0 (kh=0) or K2 (kh=1)
  const float xb = x[rc * 3 + 1];           // K1
  v2f a;
  a.x = (rm < n) ? xa : 0.0f;
  a.y = (kh == 0 && rm < n) ? xb : 0.0f;    // K3 pad = 0

  const bool full = (rowBase + 16 <= n);    // wave-uniform tail test
  const int  mBase = rowBase + (kh ? 8 : 0);  // D: lanes 16-31 hold M=r+8

#pragma unroll
  for (int tile = 0; tile < 4; ++tile) {
    const int col = tile * 16 + m;
    v2f bm;
    bm.x = v[(kh << 1) * T_THETA + col];    // row K0 or K2 of v
    bm.y = kh ? 0.0f : v[1 * T_THETA + col];

    v8f c = {};
    // 8 args: (neg_a, A, neg_b, B, c_mod, C, reuse_a, reuse_b)
    c = __builtin_amdgcn_wmma_f32_16x16x4_f32(false, a, false, bm,
                                              (short)0, c, false, false);

    float* p = nh + mBase * T_THETA + col;
    if (full) {
#pragma unroll
      for (int r = 0; r < 8; ++r) p[r * T_THETA] = c[r];
    } else {
#pragma unroll
      for (int r = 0; r < 8; ++r)
        if (mBase + r < n) p[r * T_THETA] = c[r];
    }
  }
}

// ---------------------------------------------------------------------------
// Kernel 2: zero the output (harness poisons d_out with 0xAA).
// ---------------------------------------------------------------------------
__global__ void zero_out(float* __restrict__ out, int n) {
  const int i = blockIdx.x * blockDim.x + threadIdx.x;
  if (i < n) out[i] = 0.0f;
}

// ---------------------------------------------------------------------------
// Kernel 3: fused node + edge ECT accumulation.
// grid = (B, N_CHUNK), block = 256.  Thread owns theta t = tid&63 and step
// rows s = (tid>>6) + 4*j, j=0..7 -> 256 threads cover the 32x64 accumulator
// entirely in registers.  batch[] is sorted -> binary search node range.
// Edge loop walks a shared chunk; the batch filter is block-uniform.
// ---------------------------------------------------------------------------
__device__ __forceinline__ int lower_bound_i(const int* __restrict__ a, int n, int key) {
  int lo = 0, hi = n;
  while (lo < hi) {
    const int mid = (lo + hi) >> 1;
    if (a[mid] < key) lo = mid + 1; else hi = mid;
  }
  return lo;
}

__global__ void ect_accum(const float* __restrict__ nh,
                          const int* __restrict__ esrc,
                          const int* __restrict__ edst,
                          const int* __restrict__ batch,
                          const float* __restrict__ lin,
                          float* __restrict__ out,
                          int n, int e) {
  const int b     = blockIdx.x;
  const int chunk = blockIdx.y;
  const int tid   = threadIdx.x;
  const int t     = tid & (T_THETA - 1);
  const int sBase = tid >> 6;  // 0..3

  float linv[8];
#pragma unroll
  for (int j = 0; j < 8; ++j) linv[j] = 200.0f * lin[sBase + 4 * j];

  float acc[8], eacc[8];
#pragma unroll
  for (int j = 0; j < 8; ++j) { acc[j] = 0.0f; eacc[j] = 0.0f; }

  // ---- node term: contiguous range of this graph's nodes (batch sorted) ----
  const int lo  = lower_bound_i(batch, n, b);
  const int hi  = lower_bound_i(batch, n, b + 1);
  const int cnt = hi - lo;
  const int per = (cnt + N_CHUNK - 1) / N_CHUNK;
  const int ns  = lo + chunk * per;
  const int ne  = min(hi, ns + per);
  for (int node = ns; node < ne; ++node) {
    const float hl = 200.0f * nh[node * T_THETA + t];
#pragma unroll
    for (int j = 0; j < 8; ++j)
      acc[j] += 1.0f / (1.0f + __expf(hl - linv[j]));  // sigmoid(200*(lin - h))
  }

  // ---- edge term: scan a chunk of edges, filter by graph (block-uniform) ----
  const int eper = (e + N_CHUNK - 1) / N_CHUNK;
  const int es   = chunk * eper;
  const int ee   = min(e, es + eper);
  for (int k = es; k < ee; ++k) {
    const int u = esrc[k];
    if (batch[u] != b) continue;     // uniform across the block
    const int w = edst[k];
    const float hl =
        200.0f * fmaxf(nh[u * T_THETA + t], nh[w * T_THETA + t]);
#pragma unroll
    for (int j = 0; j < 8; ++j)
      eacc[j] += 1.0f / (1.0f + __expf(hl - linv[j]));
  }

  // ---- 8 atomics per thread into out[b, s, t] ----
  float* o = out + b * (S_STEPS * T_THETA);
#pragma unroll
  for (int j = 0; j < 8; ++j)
    atomicAdd(&o[(sBase + 4 * j) * T_THETA + t], acc[j] - 0.5f * eacc[j]);
}

// ---------------------------------------------------------------------------
extern "C" void kernel_launch(void* const* d_in, const int* in_sizes, int n_in,
                              void* d_out, int out_size, void* d_ws, size_t ws_size,
                              hipStream_t stream) {
  const float* x     = (const float*)d_in[0];  // [N,3]
  const float* v     = (const float*)d_in[1];  // [3,64]
  const float* lin   = (const float*)d_in[2];  // [32]
  const int*   ei    = (const int*)d_in[3];    // [2,E]
  const int*   batch = (const int*)d_in[4];    // [N], sorted
  float*       out   = (float*)d_out;          // [B,S,T] = 65536 f32
  float*       nh    = (float*)d_ws;           // N*64 f32 scratch (5.12 MB)

  const int n = in_sizes[0] / 3;
  const int e = in_sizes[3] / 2;
  const int* esrc = ei;
  const int* edst = ei + e;

  // K1: WMMA GEMM. One wave per 16 rows.
  const int waves   = (n + 15) / 16;
  const int threads = waves * 32;
  const int blocks1 = (threads + 255) / 256;
  hipLaunchKernelGGL(nh_gemm_wmma, dim3(blocks1), dim3(256), 0, stream, x, v, nh, n);

  // K2: zero output.
  const int blocks2 = (out_size + 255) / 256;
  hipLaunchKernelGGL(zero_out, dim3(blocks2), dim3(256), 0, stream, out, out_size);

  // K3: fused node+edge ECT accumulation, 32x8 = 256 blocks.
  hipLaunchKernelGGL(ect_accum, dim3(B_GRAPH, N_CHUNK), dim3(256), 0, stream,
                     nh, esrc, edst, batch, lin, out, n, e);
}